// GNN_1571958031031
// MI455X (gfx1250) — compile-verified
//
#include <hip/hip_runtime.h>
#include <hip/hip_bf16.h>

typedef float v2f __attribute__((ext_vector_type(2)));
typedef float v8f __attribute__((ext_vector_type(8)));

#define N_NODES 100000
#define HID 128

// ---------------------------------------------------------------------------
// fp32 WMMA wrapper: D(16x16) += A(16x4) * B(4x16), full fp32 precision.
// ---------------------------------------------------------------------------
static __device__ inline v8f wmma_f32_16x16x4(v2f a, v2f b, v8f c) {
#if __has_builtin(__builtin_amdgcn_wmma_f32_16x16x4_f32)
  // 8 args: (neg_a, A, neg_b, B, c_mod, C, reuse_a, reuse_b)
  return __builtin_amdgcn_wmma_f32_16x16x4_f32(false, a, false, b, (short)0, c,
                                               false, false);
#else
  // placeholder so compilation never breaks; histogram (wmma==0) flags this
  c[0] += a[0] * b[0] + a[1] * b[1];
  return c;
#endif
}

// ---------------------------------------------------------------------------
// Degree: cnt[dst[e]] += 1.0f
// ---------------------------------------------------------------------------
__global__ __launch_bounds__(256) void degree_kernel(
    const int* __restrict__ dst, float* __restrict__ cnt, int E) {
  int e = blockIdx.x * 256 + threadIdx.x;
  if (e < E) atomicAdd(&cnt[dst[e]], 1.0f);
}

// ---------------------------------------------------------------------------
// Scatter: agg[dst[e], :] += x[src[e], :]
// F/4 threads per edge, each moves one float4 (coalesced gather, contiguous
// non-returning global_atomic_add_f32).
// ---------------------------------------------------------------------------
template <int F>
__global__ __launch_bounds__(256) void scatter_kernel(
    const float* __restrict__ x, const int* __restrict__ src,
    const int* __restrict__ dst, float* __restrict__ agg, int E) {
  constexpr int P = F / 4;  // float4 chunks per edge
  int tid = blockIdx.x * 256 + threadIdx.x;
  int total = E * P;        // <= 51.2M, fits int
  if (tid >= total) return;
  int e = tid / P;
  int p = tid - e * P;
  const float4 v = ((const float4*)(x + (size_t)src[e] * F))[p];
  float* o = agg + (size_t)dst[e] * F + p * 4;
  atomicAdd(o + 0, v.x);
  atomicAdd(o + 1, v.y);
  atomicAdd(o + 2, v.z);
  atomicAdd(o + 3, v.w);
}

// ---------------------------------------------------------------------------
// Fused mean + SAGE linear:  out = [agg/max(cnt,1) | xin] @ [Wl;Wr]^T + bl
// One block = one 16-node tile. Stage A (16 x 2F, pad +2) in LDS; wave w
// computes output columns [16w,16w+16) with v_wmma_f32_16x16x4_f32.
//
// Fragment layouts (CDNA5 ISA 7.12.2, wave32):
//   A 16x4 f32 : lane L<16 -> {A[L][k], A[L][k+1]}, lane L>=16 -> {A[L-16][k+2], A[L-16][k+3]}
//   B 4x16 f32 : lane L<16 -> {B[k][L],  B[k+1][L]}, lane L>=16 -> {B[k+2][L-16], B[k+3][L-16]}
//                (B[k][n] = Wcat[h0+n][k], so both halves read an 8B word of the
//                 row-major weight row at col k + (half?2:0))
//   C/D 16x16  : lane L<16 -> N=L, M=v ; lane L>=16 -> N=L-16, M=v+8
// ---------------------------------------------------------------------------
template <int F, bool RELU>
__global__ __launch_bounds__(256) void sage_gemm(
    const float* __restrict__ agg,  // [N, F]
    const float* __restrict__ cnt,  // [N]
    const float* __restrict__ xin,  // [N, F]
    const float* __restrict__ Wl,   // [H, F]
    const float* __restrict__ bl,   // [H]
    const float* __restrict__ Wr,   // [H, F]
    float* __restrict__ out) {      // [N, H]
  constexpr int KTOT = 2 * F;
  constexpr int STRIDE = KTOT + 2;  // even pad: 8B alignment + no bank conflicts
  __shared__ float As[16 * STRIDE];

  const int tileBase = blockIdx.x * 16;
  const int tid = threadIdx.x;

  // Stage A = [mean | x] for 16 nodes (coalesced along features).
  for (int i = tid; i < 16 * KTOT; i += 256) {
    int r = i / KTOT, c = i - r * KTOT;
    int node = tileBase + r;
    float v;
    if (c < F) {
      v = agg[(size_t)node * F + c] / fmaxf(cnt[node], 1.0f);
    } else {
      v = xin[(size_t)node * F + (c - F)];
    }
    As[r * STRIDE + c] = v;
  }
  __syncthreads();

  const int wave = tid >> 5;           // 0..7 -> output column tile
  const int lane = tid & 31;
  const int h0 = wave * 16;
  const int r = lane & 15;             // A row (M) / B column (N)
  const int koff = (lane < 16) ? 0 : 2;

  v8f acc = {};
  const float* Arow = &As[r * STRIDE];
  const float* WlRow = Wl + (size_t)(h0 + r) * F;
  const float* WrRow = Wr + (size_t)(h0 + r) * F;

  // K-loop: first F steps from Wl (aggregated part), next F from Wr (root part).
#pragma unroll 4
  for (int k = 0; k < F; k += 4) {
    v2f a = *(const v2f*)(Arow + k + koff);
    v2f b = *(const v2f*)(WlRow + k + koff);
    acc = wmma_f32_16x16x4(a, b, acc);
  }
#pragma unroll 4
  for (int k = 0; k < F; k += 4) {
    v2f a = *(const v2f*)(Arow + F + k + koff);
    v2f b = *(const v2f*)(WrRow + k + koff);
    acc = wmma_f32_16x16x4(a, b, acc);
  }

  // Epilogue: bias (+ ReLU), scatter D per the C/D VGPR layout.
  const int col = h0 + r;
  const float bias = bl[col];
  const int mbase = (lane < 16) ? 0 : 8;
#pragma unroll
  for (int v = 0; v < 8; v++) {
    float val = acc[v] + bias;
    if (RELU) val = fmaxf(val, 0.0f);
    out[(size_t)(tileBase + v + mbase) * HID + col] = val;
  }
}

// ---------------------------------------------------------------------------
// Launch: inputs = {x, edge_index, Wl1, bl1, Wr1, Wl2, bl2, Wr2}
// ---------------------------------------------------------------------------
extern "C" void kernel_launch(void* const* d_in, const int* in_sizes, int n_in,
                              void* d_out, int out_size, void* d_ws,
                              size_t ws_size, hipStream_t stream) {
  const float* x   = (const float*)d_in[0];
  const int* eidx  = (const int*)d_in[1];
  const float* Wl1 = (const float*)d_in[2];
  const float* bl1 = (const float*)d_in[3];
  const float* Wr1 = (const float*)d_in[4];
  const float* Wl2 = (const float*)d_in[5];
  const float* bl2 = (const float*)d_in[6];
  const float* Wr2 = (const float*)d_in[7];
  float* out = (float*)d_out;

  const int N = N_NODES;
  const int E = in_sizes[1] / 2;     // edge_index is [2, E]
  const int* src = eidx;
  const int* dst = eidx + E;

  // Workspace layout (floats): cnt[N] | agg1[N*64] | agg2[N*128] | h[N*128]
  float* ws   = (float*)d_ws;
  float* cnt  = ws;
  float* agg1 = ws + (size_t)N;
  float* agg2 = ws + (size_t)65 * N;
  float* h    = ws + (size_t)193 * N;

  // Zero cnt + agg1 + agg2 (first 193*N floats); h is fully overwritten.
  size_t zbytes = (size_t)193 * N * sizeof(float);
  if (zbytes > ws_size) zbytes = ws_size;
  hipMemsetAsync(d_ws, 0, zbytes, stream);

  // Degree (shared by both layers).
  degree_kernel<<<(E + 255) / 256, 256, 0, stream>>>(dst, cnt, E);

  // Layer 1: aggregate x, then h = relu([mean|x] @ [Wl1;Wr1]^T + bl1)
  {
    int total = E * (64 / 4);
    scatter_kernel<64><<<(total + 255) / 256, 256, 0, stream>>>(x, src, dst,
                                                                agg1, E);
  }
  sage_gemm<64, true><<<N / 16, 256, 0, stream>>>(agg1, cnt, x, Wl1, bl1, Wr1,
                                                  h);

  // Layer 2: aggregate h, then out = [mean|h] @ [Wl2;Wr2]^T + bl2
  {
    int total = E * (128 / 4);
    scatter_kernel<128><<<(total + 255) / 256, 256, 0, stream>>>(h, src, dst,
                                                                 agg2, E);
  }
  sage_gemm<128, false><<<N / 16, 256, 0, stream>>>(agg2, cnt, h, Wl2, bl2,
                                                    Wr2, out);
}